// PinPos_66589172957795
// MI455X (gfx1250) — compile-verified
//
#include <hip/hip_runtime.h>
#include <hip/hip_bf16.h>

// PinPos gather: out[p]            = pos[idx[p]]             + off_x[p]   (x half)
//                out[num_pins + p] = pos[num_nodes + idx[p]] + off_y[p]   (y half)
//
// Pure bandwidth-bound kernel (~96 MB traffic, ~4.2 us floor @ 23.3 TB/s).
// Strategy: b128 streaming loads/stores with TH=NT on touch-once data,
// default (cached) gathers for the L2-resident 9.6 MB pos array,
// single index load reused for both x and y gathers.

typedef float v4f __attribute__((ext_vector_type(4)));
typedef int   v4i __attribute__((ext_vector_type(4)));

__global__ __launch_bounds__(256) void PinPos_66589172957795_kernel(
    const float* __restrict__ pos,    // [2 * num_nodes] : x half then y half
    const float* __restrict__ offx,   // [num_pins]
    const float* __restrict__ offy,   // [num_pins]
    const int*   __restrict__ p2n,    // [num_pins]
    float*       __restrict__ out,    // [2 * num_pins]  : pin_x then pin_y
    int num_pins, int num_nodes)
{
    const float* __restrict__ posy = pos + num_nodes;
    int t = blockIdx.x * blockDim.x + threadIdx.x;
    int p = t * 4;

    if (p + 3 < num_pins) {
        // Streaming, touch-once traffic: non-temporal b128 loads (TH=NT)
        // so the WGP-level caches keep the reused pos[] gather lines instead.
        v4i idx = __builtin_nontemporal_load((const v4i*)(p2n  + p));
        v4f ox  = __builtin_nontemporal_load((const v4f*)(offx + p));
        v4f oy  = __builtin_nontemporal_load((const v4f*)(offy + p));

        // Gathers: default RT policy — pos (9.6 MB) is L2/WGP$ resident and
        // heavily reused across the 4M random indices.
        v4f px, py;
        px.x = pos[idx.x] + ox.x;
        px.y = pos[idx.y] + ox.y;
        px.z = pos[idx.z] + ox.z;
        px.w = pos[idx.w] + ox.w;
        py.x = posy[idx.x] + oy.x;
        py.y = posy[idx.y] + oy.y;
        py.z = posy[idx.z] + oy.z;
        py.w = posy[idx.w] + oy.w;

        // Streaming output: non-temporal b128 stores.
        __builtin_nontemporal_store(px, (v4f*)(out + p));
        __builtin_nontemporal_store(py, (v4f*)(out + num_pins + p));
    } else {
        // Scalar tail (only the last partial group of 4, if any).
        for (; p < num_pins; ++p) {
            int n = p2n[p];
            out[p]            = pos[n]  + offx[p];
            out[num_pins + p] = posy[n] + offy[p];
        }
    }
}

extern "C" void kernel_launch(void* const* d_in, const int* in_sizes, int n_in,
                              void* d_out, int out_size, void* d_ws, size_t ws_size,
                              hipStream_t stream) {
    // setup_inputs() order:
    //   0: pos                  (float32, 2*NUM_NODES)
    //   1: pin_offset_x         (float32, NUM_PINS)
    //   2: pin_offset_y         (float32, NUM_PINS)
    //   3: pin2node_map         (int32,   NUM_PINS)
    //   4: flat_node2pin_map    (int32)   -- unused by forward pin-pos
    //   5: flat_node2pin_start  (int32)   -- unused
    //   6: num_physical_nodes   (int32 scalar) -- unused
    const float* pos  = (const float*)d_in[0];
    const float* offx = (const float*)d_in[1];
    const float* offy = (const float*)d_in[2];
    const int*   p2n  = (const int*)d_in[3];
    float*       out  = (float*)d_out;

    const int num_nodes = in_sizes[0] / 2;
    const int num_pins  = in_sizes[1];

    const int pins_per_thread = 4;
    const int threads_needed  = (num_pins + pins_per_thread - 1) / pins_per_thread;
    const int block = 256;
    const int grid  = (threads_needed + block - 1) / block;

    PinPos_66589172957795_kernel<<<grid, block, 0, stream>>>(
        pos, offx, offy, p2n, out, num_pins, num_nodes);
}